// PIF_40656160424078
// MI455X (gfx1250) — compile-verified
//
#include <hip/hip_runtime.h>

typedef float v2f  __attribute__((ext_vector_type(2)));
typedef float v8f  __attribute__((ext_vector_type(8)));
typedef unsigned int u32x4 __attribute__((ext_vector_type(4)));
typedef int   i32x4 __attribute__((ext_vector_type(4)));
typedef int   i32x8 __attribute__((ext_vector_type(8)));

#define DEVC static __device__ __forceinline__

DEVC v8f wmma_f32_k4(v2f a, v2f b, v8f c) {
  // V_WMMA_F32_16X16X4_F32 : D(16x16,f32) = A(16x4,f32) x B(4x16,f32) + C
  return __builtin_amdgcn_wmma_f32_16x16x4_f32(
      /*neg_a=*/false, a, /*neg_b=*/false, b,
      /*c_mod=*/(short)0, c, /*reuse_a=*/false, /*reuse_b=*/false);
}

// Problem constants (from reference setup_inputs)
#define BB    4
#define NN    16384
#define KK    16
#define CO    32
#define CIMG  54
#define CIN1  64
#define C2PAD 72            // 70 channels padded to multiple of 4
#define NT1   (NN / 16)     // n-tiles per batch = 1024
#define WPB   4             // waves per block
#define XSTRIDE (CIN1 * 16) // one X buffer, in floats
#define EPSBN 1e-6f

// TDM: load one image tile (54 rows x 16 f32, row stride NN*KK elems) into LDS.
// All descriptor fields are wave-uniform.
DEVC void tdm_load_img_tile(const float* gaddr, unsigned lds_byte_off) {
  unsigned long long ga = (unsigned long long)(uintptr_t)gaddr;
  u32x4 g0;
  g0.x = 1u;                                        // count=1 (valid), user mode
  g0.y = lds_byte_off;                              // D#.lds_addr (bytes)
  g0.z = (unsigned)(ga & 0xFFFFFFFFull);            // global_addr[31:0]
  g0.w = (unsigned)((ga >> 32) & 0x1FFFFFFull)      // global_addr[56:32]
         | (2u << 30);                              // type=2 ("image")
  i32x8 g1;
  g1[0] = (int)(2u << 16);                          // data_size=2 -> 4B elems
  g1[1] = (int)(16u << 16);                         // tensor_dim0 = 16 (bits 63:48)
  g1[2] = (int)(54u << 16);                         // tensor_dim1 = 54 (bits 95:80)
  g1[3] = (int)(16u << 16);                         // tile_dim0 = 16 (bits 127:112)
  g1[4] = (int)54;                                  // tile_dim1 = 54 (bits 143:128)
  g1[5] = (int)(NN * KK);                           // tensor_dim0_stride = 262144
  g1[6] = 0;
  g1[7] = 0;
  i32x4 z4 = {0, 0, 0, 0};
#if __clang_major__ >= 23
  i32x8 z8 = {0, 0, 0, 0, 0, 0, 0, 0};
  __builtin_amdgcn_tensor_load_to_lds(g0, g1, z4, z4, z8, 0);
#else
  __builtin_amdgcn_tensor_load_to_lds(g0, g1, z4, z4, 0);
#endif
}

__global__ __launch_bounds__(128) void pif_fused_kernel(
    const float* __restrict__ points, const float* __restrict__ imgf,
    const float* __restrict__ dist,   const int*   __restrict__ idx,
    const float* __restrict__ w1, const float* __restrict__ b1,
    const float* __restrict__ g1, const float* __restrict__ be1,
    const float* __restrict__ m1, const float* __restrict__ v1,
    const float* __restrict__ w2, const float* __restrict__ b2,
    const float* __restrict__ g2, const float* __restrict__ be2,
    const float* __restrict__ m2, const float* __restrict__ v2,
    float* __restrict__ out)
{
  // Per-wave LDS: double-buffered X stage (2 x 64x16 f32) + pooled P (72x16 f32)
  __shared__ float lds[WPB][2 * XSTRIDE + C2PAD * 16];

  const int wave = threadIdx.x >> 5;
  const int lane = threadIdx.x & 31;
  const int lo   = lane & 15;   // N / M index within 16
  const int hf   = lane >> 4;   // lane half (K striping)

  const int tile = blockIdx.x * WPB + wave;   // 0 .. 4095
  const int b    = tile >> 10;                 // tile / NT1
  const int n0   = (tile & (NT1 - 1)) * 16;

  // LDS byte offset of the image-channel region (rows 10..63) of X buffer 0.
  // Pure value for the TDM descriptor only; all LDS *accesses* below are direct
  // GEPs on `lds` so addrspace(3) inference keeps them as DS instructions.
  const unsigned ldsImg0 = (unsigned)(uintptr_t)(&lds[wave][10 * 16]);

  // ---- preload W1 fragments (B operand of H^T = X^T * W1^T) ----
  // b[v] at lane = w1[o = lo+16*mt][c = 4s + v + 2*hf]
  v2f wf1[2][16];
#pragma unroll
  for (int mt = 0; mt < 2; ++mt) {
    const float* row = w1 + (lo + 16 * mt) * CIN1 + 2 * hf;
#pragma unroll
    for (int s = 0; s < 16; ++s) {
      wf1[mt][s].x = row[4 * s + 0];
      wf1[mt][s].y = row[4 * s + 1];
    }
  }

  // ---- fold conv bias into BN affine:  y = x*scale + shift ----
  float s1[2], sh1[2], s2[2], sh2[2];
#pragma unroll
  for (int mt = 0; mt < 2; ++mt) {
    int o = lo + 16 * mt;
    float inv1 = g1[o] * rsqrtf(v1[o] + EPSBN);
    s1[mt]  = inv1;
    sh1[mt] = be1[o] + (b1[o] - m1[o]) * inv1;
    float inv2 = g2[o] * rsqrtf(v2[o] + EPSBN);
    s2[mt]  = inv2;
    sh2[mt] = be2[o] + (b2[o] - m2[o]) * inv2;
  }

  // zero the pad rows of P (c = 70,71 for all 16 columns = 32 floats)
  lds[wave][2 * XSTRIDE + 70 * 16 + lane] = 0.0f;

  const float* ptsB  = points + b * NN * 3;
  const float* imgB  = imgf   + (size_t)b * CIMG * NN * KK;
  const float* distB = dist   + b * NN * KK;
  const int*   idxB  = idx    + b * NN * KK;

  // Prefetch image tile for j = 0 via the Tensor Data Mover
  tdm_load_img_tile(imgB + (size_t)n0 * KK, ldsImg0);

  for (int j = 0; j < 16; ++j) {
    const int n  = n0 + j;
    const int xb = (j & 1) * XSTRIDE;   // float offset of this X buffer

    // Prefetch next tile into the other buffer while this one is in flight
    if (j < 15)
      tdm_load_img_tile(imgB + (size_t)(n + 1) * KK,
                        ldsImg0 + (unsigned)(((j + 1) & 1) * XSTRIDE * 4));

    // ====== stage the non-image channels 0..9 (rows disjoint from TDM) ======
    const float ep0 = ptsB[n * 3 + 0];
    const float ep1 = ptsB[n * 3 + 1];
    const float ep2 = ptsB[n * 3 + 2];
    if (hf == 0) {
      const int k = lo;
      lds[wave][xb + 0 * 16 + k] = ep0;
      lds[wave][xb + 1 * 16 + k] = ep1;
      lds[wave][xb + 2 * 16 + k] = ep2;
      const int mIdx = idxB[n * KK + k];
      float q0 = ptsB[mIdx * 3 + 0];
      float q1 = ptsB[mIdx * 3 + 1];
      float q2 = ptsB[mIdx * 3 + 2];
      lds[wave][xb + 3 * 16 + k] = q0;
      lds[wave][xb + 4 * 16 + k] = q1;
      lds[wave][xb + 5 * 16 + k] = q2;
      lds[wave][xb + 6 * 16 + k] = ep0 - q0;
      lds[wave][xb + 7 * 16 + k] = ep1 - q1;
      lds[wave][xb + 8 * 16 + k] = ep2 - q2;
      lds[wave][xb + 9 * 16 + k] = distB[n * KK + k];
    }

    // Wait for THIS tile's TDM (in-order completion): <=1 leaves the prefetch
    // in flight; last iteration drains everything.
    __builtin_amdgcn_sched_barrier(0);
    if (j < 15) __builtin_amdgcn_s_wait_tensorcnt(1);
    else        __builtin_amdgcn_s_wait_tensorcnt(0);
    __builtin_amdgcn_sched_barrier(0);

    // ============ stage-1 GEMM: H^T[16k x 16o] = X^T * W1^T ============
    v8f acc0 = {0.f, 0.f, 0.f, 0.f, 0.f, 0.f, 0.f, 0.f};
    v8f acc1 = {0.f, 0.f, 0.f, 0.f, 0.f, 0.f, 0.f, 0.f};
#pragma unroll
    for (int s = 0; s < 16; ++s) {
      const int c0 = 4 * s + 2 * hf;
      v2f a;
      a.x = lds[wave][xb + (c0 + 0) * 16 + lo];
      a.y = lds[wave][xb + (c0 + 1) * 16 + lo];
      acc0 = wmma_f32_k4(a, wf1[0][s], acc0);
      acc1 = wmma_f32_k4(a, wf1[1][s], acc1);
    }

    // ============ BN1 + ReLU + pool over K (16) ============
    // lane holds o = lo + 16*mt, k = r + 8*hf  (r = 0..7)
    float mx[2], mn[2];
#pragma unroll
    for (int mt = 0; mt < 2; ++mt) {
      v8f acc = mt ? acc1 : acc0;
      float m  = 0.0f;   // ReLU output >= 0, so 0 is a safe max identity
      float sm = 0.0f;
#pragma unroll
      for (int r = 0; r < 8; ++r) {
        float hv = fmaxf(0.0f, acc[r] * s1[mt] + sh1[mt]);
        m  = fmaxf(m, hv);
        sm += hv;
      }
      m  = fmaxf(m, __shfl_xor(m, 16, 32));
      sm = sm + __shfl_xor(sm, 16, 32);
      mx[mt] = m;
      mn[mt] = sm * (1.0f / 16.0f);
    }

    // pooled column j: P[0..31]=max(h), P[32..34]=ep, P[35..66]=mean(h), P[67..69]=ep
    if (hf == 0) {
      float* Pw = &lds[wave][2 * XSTRIDE];
      Pw[(lo +  0) * 16 + j] = mx[0];
      Pw[(lo + 16) * 16 + j] = mx[1];
      Pw[(lo + 35) * 16 + j] = mn[0];
      Pw[(lo + 51) * 16 + j] = mn[1];
      if (lo < 3) {
        float epv = (lo == 0) ? ep0 : ((lo == 1) ? ep1 : ep2);
        Pw[(32 + lo) * 16 + j] = epv;
        Pw[(67 + lo) * 16 + j] = epv;
      }
    }
    // No barrier: buffers are wave-private; DS is in-order within a wave.
  }

  // ============ stage-2 GEMM: Out^T[16n x 16o] = P^T * W2^T ============
  v8f acc0 = {0.f, 0.f, 0.f, 0.f, 0.f, 0.f, 0.f, 0.f};
  v8f acc1 = {0.f, 0.f, 0.f, 0.f, 0.f, 0.f, 0.f, 0.f};
#pragma unroll
  for (int s = 0; s < 18; ++s) {
    const int c0 = 4 * s + 2 * hf;
    v2f a;
    a.x = lds[wave][2 * XSTRIDE + (c0 + 0) * 16 + lo];
    a.y = lds[wave][2 * XSTRIDE + (c0 + 1) * 16 + lo];
#pragma unroll
    for (int mt = 0; mt < 2; ++mt) {
      const int o = lo + 16 * mt;
      v2f bf;
      bf.x = (c0 + 0 < 70) ? w2[o * 70 + c0 + 0] : 0.0f;
      bf.y = (c0 + 1 < 70) ? w2[o * 70 + c0 + 1] : 0.0f;
      if (mt == 0) acc0 = wmma_f32_k4(a, bf, acc0);
      else         acc1 = wmma_f32_k4(a, bf, acc1);
    }
  }

  // BN2 + ReLU, transpose through LDS (reuse X buffer 0) for coalesced stores
#pragma unroll
  for (int mt = 0; mt < 2; ++mt) {
    v8f acc = mt ? acc1 : acc0;
    const int o = lo + 16 * mt;
#pragma unroll
    for (int r = 0; r < 8; ++r) {
      float ov = fmaxf(0.0f, acc[r] * s2[mt] + sh2[mt]);
      lds[wave][o * 16 + (r + 8 * hf)] = ov;   // O[o][j]
    }
  }

  float* outB = out + (size_t)(b * CO) * NN + n0;
#pragma unroll
  for (int t = 0; t < 16; ++t) {
    const int o = 2 * t + hf;          // half-wave per output row: 64B contiguous
    outB[(size_t)o * NN + lo] = lds[wave][o * 16 + lo];
  }
}

extern "C" void kernel_launch(void* const* d_in, const int* in_sizes, int n_in,
                              void* d_out, int out_size, void* d_ws, size_t ws_size,
                              hipStream_t stream) {
  const float* points = (const float*)d_in[0];
  const float* imgf   = (const float*)d_in[1];
  const float* dist   = (const float*)d_in[2];
  const int*   idx    = (const int*)  d_in[3];
  const float* w1  = (const float*)d_in[4];
  const float* b1  = (const float*)d_in[5];
  const float* g1  = (const float*)d_in[6];
  const float* be1 = (const float*)d_in[7];
  const float* m1  = (const float*)d_in[8];
  const float* v1  = (const float*)d_in[9];
  const float* w2  = (const float*)d_in[10];
  const float* b2  = (const float*)d_in[11];
  const float* g2  = (const float*)d_in[12];
  const float* be2 = (const float*)d_in[13];
  const float* m2  = (const float*)d_in[14];
  const float* v2  = (const float*)d_in[15];

  const int totalTiles = BB * NT1;             // 4096 waves of work
  const int blocks     = totalTiles / WPB;     // 1024 blocks x 128 threads

  pif_fused_kernel<<<blocks, 128, 0, stream>>>(
      points, imgf, dist, idx,
      w1, b1, g1, be1, m1, v1,
      w2, b2, g2, be2, m2, v2,
      (float*)d_out);
}